// HGN_43379169689807
// MI455X (gfx1250) — compile-verified
//
#include <hip/hip_runtime.h>
#include <math.h>

#define BLOCK 256
#define NEG_SLOPE 0.2f

typedef float v2f __attribute__((ext_vector_type(2)));
typedef float v8f __attribute__((ext_vector_type(8)));

static inline int cdiv_i(long long a, long long b) { return (int)((a + b - 1) / b); }

// ---------------------------------------------------------------------------
// utility
// ---------------------------------------------------------------------------
__global__ void HGN_fill(float* __restrict__ p, int n, float v) {
    int i = blockIdx.x * blockDim.x + threadIdx.x;
    if (i < n) p[i] = v;
}

__device__ __forceinline__ void HGN_edge_sd(const int* __restrict__ ei, int E, int e,
                                            int& src, int& dst) {
    if (e < E) { src = ei[e]; dst = ei[E + e]; }
    else       { src = e - E; dst = e - E; }
}

__device__ __forceinline__ float HGN_leaky(float x) {
    return x >= 0.0f ? x : NEG_SLOPE * x;
}

// ---------------------------------------------------------------------------
// WMMA f32 GEMM: C[N,64] = A[N,K] * B   (B is [K,64] row-major, or B^T of a
// [64,K]-row-major matrix when TRANSB). One wave computes a 16x64 stripe
// using V_WMMA_F32_16X16X4_F32 (exact f32 math).
// Fragment layouts per CDNA5 ISA 7.12.2:
//   A 16x4:  lane m = lane&15 ; VGPR r holds K = (lane>>4)*2 + r
//   B 4x16:  lane n = lane&15 ; VGPR r holds K = (lane>>4)*2 + r
//   C 16x16: lane n = lane&15 ; VGPR r holds M = r + (lane>>4)*8
// Epilogue: wave-uniform fast path for full tiles (unpredicated stores);
// only the single boundary wave takes the per-row predicated path.
// ---------------------------------------------------------------------------
template <int K, bool TRANSB>
__global__ void HGN_gemm64(const float* __restrict__ A, const float* __restrict__ B,
                           float* __restrict__ C, int nRows) {
    const int lane = threadIdx.x & 31;
    const int wave = threadIdx.x >> 5;
    const int rowBase = (blockIdx.x * (BLOCK / 32) + wave) * 16;
    if (rowBase >= nRows) return;              // wave-uniform: EXEC stays full
    const int m   = lane & 15;
    const int hi2 = (lane >> 4) * 2;           // 0 or 2
    int arow = rowBase + m;
    if (arow >= nRows) arow = nRows - 1;       // clamp loads, keep EXEC full

    v8f acc[4];
#pragma unroll
    for (int t = 0; t < 4; ++t)
#pragma unroll
        for (int r = 0; r < 8; ++r) acc[t][r] = 0.0f;

#pragma unroll
    for (int k0 = 0; k0 < K; k0 += 4) {
        v2f a;
        a.x = A[arow * K + k0 + hi2];
        a.y = A[arow * K + k0 + hi2 + 1];
#pragma unroll
        for (int t = 0; t < 4; ++t) {
            const int n = t * 16 + m;
            v2f b;
            if (TRANSB) {
                b.x = B[n * K + k0 + hi2];
                b.y = B[n * K + k0 + hi2 + 1];
            } else {
                b.x = B[(k0 + hi2) * 64 + n];
                b.y = B[(k0 + hi2 + 1) * 64 + n];
            }
            acc[t] = __builtin_amdgcn_wmma_f32_16x16x4_f32(
                false, a, false, b, (short)0, acc[t], false, false);
        }
    }

    const int rOff = (lane >> 4) * 8;
    if (rowBase + 16 <= nRows) {
        // full tile: no predication (wave-uniform branch)
#pragma unroll
        for (int t = 0; t < 4; ++t)
#pragma unroll
            for (int r = 0; r < 8; ++r)
                C[(rowBase + rOff + r) * 64 + t * 16 + m] = acc[t][r];
    } else {
        // boundary tile (at most one wave in the grid)
#pragma unroll
        for (int t = 0; t < 4; ++t)
#pragma unroll
            for (int r = 0; r < 8; ++r) {
                int row = rowBase + rOff + r;
                if (row < nRows) C[row * 64 + t * 16 + m] = acc[t][r];
            }
    }
}

// ---------------------------------------------------------------------------
// per-node attention scores: s = h . a_src, d = h . a_dst
// ---------------------------------------------------------------------------
template <int F>
__global__ void HGN_scores(const float* __restrict__ h, const float* __restrict__ as,
                           const float* __restrict__ ad,
                           float* __restrict__ s, float* __restrict__ d, int n) {
    int i = blockIdx.x * blockDim.x + threadIdx.x;
    if (i >= n) return;
    float ss = 0.0f, dd = 0.0f;
#pragma unroll
    for (int f = 0; f < F; ++f) {
        float hv = h[i * F + f];
        ss += hv * as[f];
        dd += hv * ad[f];
    }
    s[i] = ss;
    d[i] = dd;
}

// ---------------------------------------------------------------------------
// softmax pieces over incoming edges (ordered-uint float atomicMax trick)
// ---------------------------------------------------------------------------
__global__ void HGN_edge_max(const int* __restrict__ ei, int E, int Etot,
                             const float* __restrict__ s, const float* __restrict__ d,
                             unsigned* __restrict__ mkey) {
    int e = blockIdx.x * blockDim.x + threadIdx.x;
    if (e >= Etot) return;
    int src, dst;
    HGN_edge_sd(ei, E, e, src, dst);
    float v = HGN_leaky(s[src] + d[dst]);
    unsigned u = __float_as_uint(v);
    unsigned key = (u & 0x80000000u) ? ~u : (u | 0x80000000u);
    atomicMax(&mkey[dst], key);
}

__global__ void HGN_decode_max(unsigned* __restrict__ mkey, int n) {
    int i = blockIdx.x * blockDim.x + threadIdx.x;
    if (i >= n) return;
    unsigned key = mkey[i];
    unsigned u = (key & 0x80000000u) ? (key & 0x7fffffffu) : ~key;
    ((float*)mkey)[i] = __uint_as_float(u);
}

__global__ void HGN_edge_expsum(const int* __restrict__ ei, int E, int Etot,
                                const float* __restrict__ s, const float* __restrict__ d,
                                const float* __restrict__ m, float* __restrict__ den) {
    int e = blockIdx.x * blockDim.x + threadIdx.x;
    if (e >= Etot) return;
    int src, dst;
    HGN_edge_sd(ei, E, e, src, dst);
    float ex = expf(HGN_leaky(s[src] + d[dst]) - m[dst]);
    atomicAdd(&den[dst], ex);
}

// agg[dst] += alpha * h[src], TPE threads cooperate per edge
template <int F, int TPE>
__global__ void HGN_edge_agg(const int* __restrict__ ei, int E, int Etot,
                             const float* __restrict__ s, const float* __restrict__ d,
                             const float* __restrict__ m, const float* __restrict__ den,
                             const float* __restrict__ h, float* __restrict__ agg) {
    int gid = blockIdx.x * blockDim.x + threadIdx.x;
    int e = gid / TPE;
    int part = gid % TPE;
    if (e >= Etot) return;
    int src, dst;
    HGN_edge_sd(ei, E, e, src, dst);
    float alpha = expf(HGN_leaky(s[src] + d[dst]) - m[dst]) / den[dst];
    const int CH = F / TPE;
#pragma unroll
    for (int f = 0; f < CH; ++f) {
        int col = part * CH + f;
        atomicAdd(&agg[dst * F + col], alpha * h[src * F + col]);
    }
}

template <int F>
__global__ void HGN_bias_tanh(const float* __restrict__ agg, const float* __restrict__ b,
                              float* __restrict__ t, int n) {
    int i = blockIdx.x * blockDim.x + threadIdx.x;
    if (i >= n * F) return;
    int f = i % F;
    t[i] = tanhf(agg[i] + b[f]);
}

// ---------------------------------------------------------------------------
// backward pieces
// ---------------------------------------------------------------------------
template <int F>
__global__ void HGN_nodeT(const float* __restrict__ g, const float* __restrict__ agg,
                          float* __restrict__ T, int n) {
    int i = blockIdx.x * blockDim.x + threadIdx.x;
    if (i >= n) return;
    float acc = 0.0f;
#pragma unroll
    for (int f = 0; f < F; ++f) acc += g[i * F + f] * agg[i * F + f];
    T[i] = acc;
}

// layer-3 special: upstream grad g == 1 everywhere (energy = sum(y))
__global__ void HGN_T3(const float* __restrict__ agg3, float* __restrict__ T, int n) {
    int i = blockIdx.x * blockDim.x + threadIdx.x;
    if (i >= n) return;
    T[i] = agg3[i * 2] + agg3[i * 2 + 1];
}

template <int F>
__global__ void HGN_edge_bwd(const int* __restrict__ ei, int E, int Etot,
                             const float* __restrict__ s, const float* __restrict__ d,
                             const float* __restrict__ m, const float* __restrict__ den,
                             const float* __restrict__ h, const float* __restrict__ g,
                             const float* __restrict__ T,
                             float* __restrict__ gs, float* __restrict__ gd,
                             float* __restrict__ gh) {
    int e = blockIdx.x * blockDim.x + threadIdx.x;
    if (e >= Etot) return;
    int src, dst;
    HGN_edge_sd(ei, E, e, src, dst);
    float pre = s[src] + d[dst];
    float alpha = expf(HGN_leaky(pre) - m[dst]) / den[dst];
    float dot = 0.0f;
#pragma unroll
    for (int f = 0; f < F; ++f) dot += g[dst * F + f] * h[src * F + f];
    float de = alpha * (dot - T[dst]);
    float dpre = de * (pre >= 0.0f ? 1.0f : NEG_SLOPE);
    atomicAdd(&gs[src], dpre);
    atomicAdd(&gd[dst], dpre);
#pragma unroll
    for (int f = 0; f < F; ++f) atomicAdd(&gh[src * F + f], alpha * g[dst * F + f]);
}

__global__ void HGN_edge_bwd3(const int* __restrict__ ei, int E, int Etot,
                              const float* __restrict__ s, const float* __restrict__ d,
                              const float* __restrict__ m, const float* __restrict__ den,
                              const float* __restrict__ h3, const float* __restrict__ T,
                              float* __restrict__ gs, float* __restrict__ gd,
                              float* __restrict__ gh3) {
    int e = blockIdx.x * blockDim.x + threadIdx.x;
    if (e >= Etot) return;
    int src, dst;
    HGN_edge_sd(ei, E, e, src, dst);
    float pre = s[src] + d[dst];
    float alpha = expf(HGN_leaky(pre) - m[dst]) / den[dst];
    float dot = h3[src * 2] + h3[src * 2 + 1];     // g == 1
    float de = alpha * (dot - T[dst]);
    float dpre = de * (pre >= 0.0f ? 1.0f : NEG_SLOPE);
    atomicAdd(&gs[src], dpre);
    atomicAdd(&gd[dst], dpre);
    atomicAdd(&gh3[src * 2], alpha);
    atomicAdd(&gh3[src * 2 + 1], alpha);
}

template <int F>
__global__ void HGN_gh_finish(float* __restrict__ gh, const float* __restrict__ gs,
                              const float* __restrict__ gd, const float* __restrict__ as,
                              const float* __restrict__ ad, int n) {
    int i = blockIdx.x * blockDim.x + threadIdx.x;
    if (i >= n * F) return;
    int node = i / F, f = i % F;
    gh[i] += gs[node] * as[f] + gd[node] * ad[f];
}

// layer-3 forward GEMM (HOUT=2, tiny): h3 = t2 @ W3
__global__ void HGN_gemm_h3(const float* __restrict__ t2, const float* __restrict__ W3,
                            float* __restrict__ h3, int n) {
    int i = blockIdx.x * blockDim.x + threadIdx.x;
    if (i >= n) return;
    float a0 = 0.0f, a1 = 0.0f;
#pragma unroll
    for (int f = 0; f < 64; ++f) {
        float v = t2[i * 64 + f];
        a0 += v * W3[f * 2];
        a1 += v * W3[f * 2 + 1];
    }
    h3[i * 2] = a0;
    h3[i * 2 + 1] = a1;
}

// g_out2 = (gh3 @ W3^T) * (1 - t2^2)
__global__ void HGN_gt2(const float* __restrict__ gh3, const float* __restrict__ W3,
                        const float* __restrict__ t2, float* __restrict__ g2, int n) {
    int i = blockIdx.x * blockDim.x + threadIdx.x;
    if (i >= n * 64) return;
    int node = i / 64, f = i % 64;
    float tv = t2[i];
    g2[i] = (gh3[node * 2] * W3[f * 2] + gh3[node * 2 + 1] * W3[f * 2 + 1]) * (1.0f - tv * tv);
}

__global__ void HGN_tanh_bwd(float* __restrict__ g, const float* __restrict__ t, int n) {
    int i = blockIdx.x * blockDim.x + threadIdx.x;
    if (i >= n) return;
    float tv = t[i];
    g[i] *= (1.0f - tv * tv);
}

// gx1 = gh1 @ W1^T (4 cols), then out = gx1 @ J^T = [gx2, gx3, -gx0, -gx1]
__global__ void HGN_final(const float* __restrict__ gh1, const float* __restrict__ W1,
                          float* __restrict__ out, int n) {
    int i = blockIdx.x * blockDim.x + threadIdx.x;
    if (i >= n) return;
    float gx0 = 0.0f, gx1 = 0.0f, gx2 = 0.0f, gx3 = 0.0f;
#pragma unroll
    for (int f = 0; f < 64; ++f) {
        float gv = gh1[i * 64 + f];
        gx0 += gv * W1[0 * 64 + f];
        gx1 += gv * W1[1 * 64 + f];
        gx2 += gv * W1[2 * 64 + f];
        gx3 += gv * W1[3 * 64 + f];
    }
    out[i * 4 + 0] = gx2;
    out[i * 4 + 1] = gx3;
    out[i * 4 + 2] = -gx0;
    out[i * 4 + 3] = -gx1;
}

// ---------------------------------------------------------------------------
// host orchestration
// ---------------------------------------------------------------------------
extern "C" void kernel_launch(void* const* d_in, const int* in_sizes, int n_in,
                              void* d_out, int out_size, void* d_ws, size_t ws_size,
                              hipStream_t stream) {
    const float* x   = (const float*)d_in[0];
    const int*   ei  = (const int*)d_in[1];
    const float* W1  = (const float*)d_in[2];
    const float* as1 = (const float*)d_in[3];
    const float* ad1 = (const float*)d_in[4];
    const float* b1  = (const float*)d_in[5];
    const float* W2  = (const float*)d_in[6];
    const float* as2 = (const float*)d_in[7];
    const float* ad2 = (const float*)d_in[8];
    const float* b2  = (const float*)d_in[9];
    const float* W3  = (const float*)d_in[10];
    const float* as3 = (const float*)d_in[11];
    const float* ad3 = (const float*)d_in[12];
    // b3 (d_in[13]) has zero gradient path to x; unused.

    const int N = in_sizes[0] / 4;
    const int E = in_sizes[1] / 2;
    const int Etot = E + N;
    const int N64 = N * 64;
    const int N2  = N * 2;

    float* w = (float*)d_ws;
    long long off = 0;
    auto alloc = [&](long long cnt) { float* p = w + off; off += cnt; return p; };
    float *h1 = alloc(N64), *t1 = alloc(N64), *agg1 = alloc(N64);
    float *h2 = alloc(N64), *t2 = alloc(N64), *agg2 = alloc(N64);
    float *gh = alloc(N64), *gbuf = alloc(N64);
    float *h3 = alloc(N2), *agg3 = alloc(N2), *gh3 = alloc(N2);
    float *s1 = alloc(N), *d1 = alloc(N), *m1 = alloc(N), *den1 = alloc(N);
    float *s2 = alloc(N), *d2 = alloc(N), *m2 = alloc(N), *den2 = alloc(N);
    float *s3 = alloc(N), *d3 = alloc(N), *m3 = alloc(N), *den3 = alloc(N);
    float *Tb = alloc(N), *gs = alloc(N), *gd = alloc(N);
    (void)ws_size; (void)n_in; (void)out_size;

    const dim3 blk(BLOCK);
    const int gN    = cdiv_i(N, BLOCK);
    const int gN2   = cdiv_i(N2, BLOCK);
    const int gN64  = cdiv_i(N64, BLOCK);
    const int gE    = cdiv_i(Etot, BLOCK);
    const int gE8   = cdiv_i((long long)Etot * 8, BLOCK);
    const int gGemm = cdiv_i(N, (BLOCK / 32) * 16);   // 128 rows / block

    // ---------------- forward layer 1 ----------------
    HGN_gemm64<4, false><<<gGemm, blk, 0, stream>>>(x, W1, h1, N);
    HGN_scores<64><<<gN, blk, 0, stream>>>(h1, as1, ad1, s1, d1, N);
    HGN_fill<<<gN, blk, 0, stream>>>(m1, N, 0.0f);     // ordered-key 0 == -inf
    HGN_fill<<<gN, blk, 0, stream>>>(den1, N, 0.0f);
    HGN_fill<<<gN64, blk, 0, stream>>>(agg1, N64, 0.0f);
    HGN_edge_max<<<gE, blk, 0, stream>>>(ei, E, Etot, s1, d1, (unsigned*)m1);
    HGN_decode_max<<<gN, blk, 0, stream>>>((unsigned*)m1, N);
    HGN_edge_expsum<<<gE, blk, 0, stream>>>(ei, E, Etot, s1, d1, m1, den1);
    HGN_edge_agg<64, 8><<<gE8, blk, 0, stream>>>(ei, E, Etot, s1, d1, m1, den1, h1, agg1);
    HGN_bias_tanh<64><<<gN64, blk, 0, stream>>>(agg1, b1, t1, N);

    // ---------------- forward layer 2 ----------------
    HGN_gemm64<64, false><<<gGemm, blk, 0, stream>>>(t1, W2, h2, N);
    HGN_scores<64><<<gN, blk, 0, stream>>>(h2, as2, ad2, s2, d2, N);
    HGN_fill<<<gN, blk, 0, stream>>>(m2, N, 0.0f);
    HGN_fill<<<gN, blk, 0, stream>>>(den2, N, 0.0f);
    HGN_fill<<<gN64, blk, 0, stream>>>(agg2, N64, 0.0f);
    HGN_edge_max<<<gE, blk, 0, stream>>>(ei, E, Etot, s2, d2, (unsigned*)m2);
    HGN_decode_max<<<gN, blk, 0, stream>>>((unsigned*)m2, N);
    HGN_edge_expsum<<<gE, blk, 0, stream>>>(ei, E, Etot, s2, d2, m2, den2);
    HGN_edge_agg<64, 8><<<gE8, blk, 0, stream>>>(ei, E, Etot, s2, d2, m2, den2, h2, agg2);
    HGN_bias_tanh<64><<<gN64, blk, 0, stream>>>(agg2, b2, t2, N);

    // ---------------- forward layer 3 ----------------
    HGN_gemm_h3<<<gN, blk, 0, stream>>>(t2, W3, h3, N);
    HGN_scores<2><<<gN, blk, 0, stream>>>(h3, as3, ad3, s3, d3, N);
    HGN_fill<<<gN, blk, 0, stream>>>(m3, N, 0.0f);
    HGN_fill<<<gN, blk, 0, stream>>>(den3, N, 0.0f);
    HGN_fill<<<gN2, blk, 0, stream>>>(agg3, N2, 0.0f);
    HGN_edge_max<<<gE, blk, 0, stream>>>(ei, E, Etot, s3, d3, (unsigned*)m3);
    HGN_decode_max<<<gN, blk, 0, stream>>>((unsigned*)m3, N);
    HGN_edge_expsum<<<gE, blk, 0, stream>>>(ei, E, Etot, s3, d3, m3, den3);
    HGN_edge_agg<2, 1><<<gE, blk, 0, stream>>>(ei, E, Etot, s3, d3, m3, den3, h3, agg3);

    // ---------------- backward layer 3 (g == 1) ----------------
    HGN_T3<<<gN, blk, 0, stream>>>(agg3, Tb, N);
    HGN_fill<<<gN, blk, 0, stream>>>(gs, N, 0.0f);
    HGN_fill<<<gN, blk, 0, stream>>>(gd, N, 0.0f);
    HGN_fill<<<gN2, blk, 0, stream>>>(gh3, N2, 0.0f);
    HGN_edge_bwd3<<<gE, blk, 0, stream>>>(ei, E, Etot, s3, d3, m3, den3, h3, Tb, gs, gd, gh3);
    HGN_gh_finish<2><<<gN2, blk, 0, stream>>>(gh3, gs, gd, as3, ad3, N);
    HGN_gt2<<<gN64, blk, 0, stream>>>(gh3, W3, t2, gbuf, N);   // gbuf = dL/d out2

    // ---------------- backward layer 2 ----------------
    HGN_nodeT<64><<<gN, blk, 0, stream>>>(gbuf, agg2, Tb, N);
    HGN_fill<<<gN, blk, 0, stream>>>(gs, N, 0.0f);
    HGN_fill<<<gN, blk, 0, stream>>>(gd, N, 0.0f);
    HGN_fill<<<gN64, blk, 0, stream>>>(gh, N64, 0.0f);
    HGN_edge_bwd<64><<<gE, blk, 0, stream>>>(ei, E, Etot, s2, d2, m2, den2, h2, gbuf, Tb,
                                             gs, gd, gh);
    HGN_gh_finish<64><<<gN64, blk, 0, stream>>>(gh, gs, gd, as2, ad2, N);
    HGN_gemm64<64, true><<<gGemm, blk, 0, stream>>>(gh, W2, gbuf, N);  // gbuf = dL/d t1
    HGN_tanh_bwd<<<gN64, blk, 0, stream>>>(gbuf, t1, N64);             // gbuf = dL/d out1

    // ---------------- backward layer 1 ----------------
    HGN_nodeT<64><<<gN, blk, 0, stream>>>(gbuf, agg1, Tb, N);
    HGN_fill<<<gN, blk, 0, stream>>>(gs, N, 0.0f);
    HGN_fill<<<gN, blk, 0, stream>>>(gd, N, 0.0f);
    HGN_fill<<<gN64, blk, 0, stream>>>(gh, N64, 0.0f);
    HGN_edge_bwd<64><<<gE, blk, 0, stream>>>(ei, E, Etot, s1, d1, m1, den1, h1, gbuf, Tb,
                                             gs, gd, gh);
    HGN_gh_finish<64><<<gN64, blk, 0, stream>>>(gh, gs, gd, as1, ad1, N);

    // gx1 = gh @ W1^T ; out = gx1 @ J^T
    HGN_final<<<gN, blk, 0, stream>>>(gh, W1, (float*)d_out, N);
}